// FeatureTransformerSlice_5428838662248
// MI455X (gfx1250) — compile-verified
//
#include <hip/hip_runtime.h>

// ---------------------------------------------------------------------------
// Sparse feature transformer: out[b,:] = bias + sum_k val[b,k] * W[idx[b,k],:]
// B=4096, K<=32 (ragged, -1 sentinel suffix), NUM_INPUTS=45056, O=1024, f32.
//
// MI455X strategy (memory-latency-bound gather; weight table ~176MiB fits L2):
//  * One workgroup (256 threads = 8 waves) per batch sample.
//  * Wave 0 issues TDM gather-mode tensor_load_to_lds: 16-bit row indices
//    (45056 < 2^16) gather up to 16 weight rows (4 KB each) per descriptor
//    into LDS; two descriptors cover K<=32 (TENSORcnt + workgroup barrier).
//    The DMA engine owns the random 4KB row fetches, decoupled from waves.
//  * Contraction with V_WMMA_F32_16X16X4_F32: A (16x4) carries the sample's
//    4 feature values replicated down all 16 M rows, B (4x16) is 4 gathered
//    rows x 16 output columns from LDS; every row of D equals the desired
//    16-wide partial output. K-chunk outer loop, 8 column tiles inner with
//    8 live accumulators; values come from a broadcast ds_load_b128 of a
//    zero-padded LDS staging array (no scalar branches in the hot loop).
//  * Lanes 0..15 (M=0 of the D layout) store D + bias (bias prefetched
//    before the barrier to overlap with the TDM gather).
// ---------------------------------------------------------------------------

typedef float v2f  __attribute__((ext_vector_type(2)));
typedef float v4f  __attribute__((ext_vector_type(4)));
typedef float v8f  __attribute__((ext_vector_type(8)));
typedef unsigned int u32x4 __attribute__((ext_vector_type(4)));
typedef int   i32x8 __attribute__((ext_vector_type(8)));
typedef int   i32x4 __attribute__((ext_vector_type(4)));

#define FT_NUM_INPUTS  45056
#define FT_NUM_OUTPUTS 1024
#define FT_MAX_ACTIVE  32

__global__ __launch_bounds__(256)
void ft_tdm_wmma_kernel(const int*   __restrict__ fidx,
                        const float* __restrict__ fval,
                        const float* __restrict__ weight,
                        const float* __restrict__ bias,
                        float*       __restrict__ out)
{
    // Gathered weight rows for this sample: 32 rows x 1024 f32 = 128 KB LDS.
    __shared__ float smem[FT_MAX_ACTIVE * FT_NUM_OUTPUTS];
    __shared__ __align__(16) float svals[FT_MAX_ACTIVE];   // zero-padded values

    const int b    = blockIdx.x;
    const int lane = threadIdx.x & 31;
    const int wave = threadIdx.x >> 5;
    const int base = b * FT_MAX_ACTIVE;

    // ---- Block-uniform scalar loads of this sample's indices (for count). --
    int idxs[FT_MAX_ACTIVE];
    #pragma unroll
    for (int k = 0; k < FT_MAX_ACTIVE; ++k)
        idxs[k] = fidx[base + k];

    int count = 0;
    #pragma unroll
    for (int k = 0; k < FT_MAX_ACTIVE; ++k)
        count += (idxs[k] >= 0) ? 1 : 0;          // sentinels are a suffix

    const int kpad = (count + 3) & ~3;            // round K up to WMMA chunk

    // Pad unused gather slots with row 0 (valid memory; weighted by 0.0).
    #pragma unroll
    for (int k = 0; k < FT_MAX_ACTIVE; ++k)
        if (k >= count) idxs[k] = 0;

    // ---- Wave 0: stage padded values in LDS + issue TDM gathers. ----------
    if (wave == 0) {
        svals[lane] = (lane < count) ? fval[base + lane] : 0.0f;

        const unsigned long long wbase = (unsigned long long)(size_t)weight;
        const unsigned int ldsbase = (unsigned int)(size_t)(&smem[0]); // low 32b = LDS addr

        #pragma unroll
        for (int g = 0; g < 2; ++g) {
            const int row0 = g * 16;
            if (row0 < kpad) {
                const int nv = (kpad - row0 < 16) ? (kpad - row0) : 16;

                u32x4 g0; i32x8 g1; i32x4 g2, g3; i32x8 g23;
                // --- D# group 0: count=1, gather_mode=1, 16-bit indices ---
                g0[0] = 0x80000001u;                       // [0]=count, [31]=gather_mode
                g0[1] = ldsbase + (unsigned)row0 * FT_NUM_OUTPUTS * 4u;   // lds_addr
                g0[2] = (unsigned)(wbase & 0xffffffffu);                  // global_addr lo
                g0[3] = (unsigned)((wbase >> 32) & 0x1ffffffu) | (2u << 30); // addr hi | type=2

                // --- D# group 1 ---
                g1[0] = (2 << 16);                         // data_size = 4 bytes, mask=0
                g1[1] = (FT_NUM_OUTPUTS << 16);            // tensor_dim0[15:0] @ bit48
                g1[2] = (int)(((unsigned)FT_NUM_INPUTS & 0xffffu) << 16); // tensor_dim1 lo @ bit80
                g1[3] = (FT_NUM_OUTPUTS << 16);            // tile_dim0 @ bit112
                g1[4] = nv;                                // tile_dim1 = #valid gather indices
                g1[5] = FT_NUM_OUTPUTS;                    // tensor_dim0_stride[31:0]
                g1[6] = 0;
                g1[7] = 0;

                // --- D# groups 2/3: sixteen 16-bit row indices ---
                #pragma unroll
                for (int j = 0; j < 4; ++j) {
                    g2[j] = (idxs[row0 + 2*j]     & 0xffff) |
                            ((idxs[row0 + 2*j + 1] & 0xffff) << 16);
                    g3[j] = (idxs[row0 + 8 + 2*j]     & 0xffff) |
                            ((idxs[row0 + 8 + 2*j + 1] & 0xffff) << 16);
                }
                // Trailing int32x8 operand of the 6-arg builtin carries the
                // concatenated group2|group3 index payload as well.
                #pragma unroll
                for (int j = 0; j < 4; ++j) { g23[j] = g2[j]; g23[j + 4] = g3[j]; }

                __builtin_amdgcn_tensor_load_to_lds(g0, g1, g2, g3, g23, 0);
            }
        }
        __builtin_amdgcn_s_wait_tensorcnt(0);
    }

    // Prefetch per-tile bias while the TDM gather is in flight.
    const bool lo   = lane < 16;
    const int  nsel = lane & 15;
    const int  rsel = lo ? 0 : 2;     // K-half owned by this lane group

    float bs[8];
    #pragma unroll
    for (int t = 0; t < 8; ++t)
        bs[t] = bias[(wave * 8 + t) * 16 + nsel];

    __syncthreads();   // LDS now holds kpad gathered rows + padded values

    // ---- WMMA contraction: D = A(16x4, values replicated) x B(4x16 rows). --
    v8f C[8];
    #pragma unroll
    for (int t = 0; t < 8; ++t) C[t] = (v8f){};

    for (int kc = 0; kc < kpad; kc += 4) {
        // Broadcast ds_load_b128 of 4 padded values (uniform address).
        const v4f vv = *reinterpret_cast<const v4f*>(&svals[kc]);

        // A 16x4 f32 layout: VGPR0 = K0|K2 halves, VGPR1 = K1|K3 halves;
        // identical down all 16 M rows.
        v2f A;
        A.x = lo ? vv.x : vv.z;
        A.y = lo ? vv.y : vv.w;

        const int r0 = kc + rsel;
        #pragma unroll
        for (int t = 0; t < 8; ++t) {
            const int col0 = (wave * 8 + t) * 16;   // 8 waves x 8 tiles x 16 = 1024
            // B 4x16 f32: row k' striped across lanes within a VGPR,
            // halves split K by 2 (mirror of the A layout).
            v2f B;
            B.x = smem[ r0      * FT_NUM_OUTPUTS + col0 + nsel];
            B.y = smem[(r0 + 1) * FT_NUM_OUTPUTS + col0 + nsel];

            C[t] = __builtin_amdgcn_wmma_f32_16x16x4_f32(
                       /*neg_a=*/false, A, /*neg_b=*/false, B,
                       /*c_mod=*/(short)0, C[t], /*reuse_a=*/false, /*reuse_b=*/false);
        }
    }

    // Every D row is identical; take M=0 (VGPR 0, lanes 0..15).
    if (lo) {
        #pragma unroll
        for (int t = 0; t < 8; ++t) {
            const int col0 = (wave * 8 + t) * 16;
            out[b * FT_NUM_OUTPUTS + col0 + nsel] = C[t][0] + bs[t];
        }
    }
}

extern "C" void kernel_launch(void* const* d_in, const int* in_sizes, int n_in,
                              void* d_out, int out_size, void* d_ws, size_t ws_size,
                              hipStream_t stream)
{
    const int*   fidx   = (const int*)  d_in[0];
    const float* fval   = (const float*)d_in[1];
    const float* weight = (const float*)d_in[2];
    const float* bias   = (const float*)d_in[3];
    float*       out    = (float*)d_out;

    const int batch = in_sizes[0] / FT_MAX_ACTIVE;   // 4096

    ft_tdm_wmma_kernel<<<dim3(batch), dim3(256), 0, stream>>>(
        fidx, fval, weight, bias, out);
}